// YOLOLayer_19739669693143
// MI455X (gfx1250) — compile-verified
//
#include <hip/hip_runtime.h>
#include <stdint.h>

// YOLO detection head: (64, 3*85, 76, 76) f32 -> (64, 3*76*76, 85) f32
// Memory-bound transpose + elementwise sigmoid/exp. ~754MB traffic total
// (~32us floor at 23.3 TB/s). Transpose is performed by the CDNA5 async
// global->LDS DMA via per-lane LDS addresses; phase 2 maps one wave per
// output pixel so channel special-casing is lane-structured, leaving the
// bulk path at the irreducible 4 VALU/element sigmoid cost.

#define G        76          // grid size
#define NA       3           // anchors
#define CH       85          // 5 + 80 classes
#define GG       (G * G)     // 5776
#define BLOCK    256         // 8 wave32s
#define NWAVE    (BLOCK / 32)

__device__ __forceinline__ float sigmoidf_fast(float v) {
    // 1 / (1 + exp(-v)) with native exp/rcp
    return __builtin_amdgcn_rcpf(1.0f + __expf(-v));
}

__global__ __launch_bounds__(BLOCK) void yolo_head_kernel(
    const float* __restrict__ x,
    const int*   __restrict__ img_dim_p,
    float*       __restrict__ out)
{
    // One block per (batch, anchor, row y). Tile = 85 channels x 76 pixels.
    // LDS holds the tile ALREADY TRANSPOSED to output order: smem[s*CH + k].
    __shared__ __align__(16) float smem[G * CH];   // 25840 B

    const int bid = blockIdx.x;
    const int y   = bid % G;
    const int ba  = bid / G;
    const int a   = ba % NA;
    const int b   = ba / NA;

    // ---------------- Phase 1: transposing async DMA global -> LDS ----------
    // Global side: lanes walk channel-major input (consecutive xcol) -> each
    // wave issues 128B contiguous reads. LDS side: scatter to [s][k] order;
    // lane stride = 85 words -> bank stride 21, gcd(85,64)=1 -> conflict-free.
    const float*   src      = x + ((size_t)(b * (NA * CH) + a * CH) * GG + (size_t)y * G);
    const uint32_t lds_base = (uint32_t)(uintptr_t)(&smem[0]);

    for (int j = threadIdx.x; j < CH * G; j += BLOCK) {
        const int ch   = j / G;
        const int xcol = j - ch * G;
        const float*   gptr  = src + (size_t)ch * GG + xcol;
        const uint32_t laddr = lds_base + (uint32_t)((xcol * CH + ch) * 4);
        asm volatile("global_load_async_to_lds_b32 %0, %1, off th:TH_LOAD_NT"
                     :: "v"(laddr), "v"(gptr)
                     : "memory");
    }
    // Wait for this wave's async DMA (ASYNCcnt), then cross-wave barrier.
    asm volatile("s_wait_asynccnt 0x0" ::: "memory");
    __syncthreads();

    // ---------------- Phase 2: one wave per output pixel s ------------------
    const float stride   = (float)img_dim_p[0] / (float)G;           // 8.0
    const float anchor_w = (a == 0 ? 116.0f : (a == 1 ? 156.0f : 373.0f)) / stride;
    const float anchor_h = (a == 0 ?  90.0f : (a == 1 ? 198.0f : 326.0f)) / stride;

    float* outp = out + ((size_t)(b * NA + a) * GG + (size_t)y * G) * CH;
    const float fy = (float)y;

    const int lane = threadIdx.x & 31;
    const int wave = threadIdx.x >> 5;

    for (int s = wave; s < G; s += NWAVE) {
        const float* row  = &smem[s * CH];
        float*       orow = outp + s * CH;

        // Chunk 0: k = lane (0..31). Lanes 0-3 carry the box special cases.
        {
            const float v = row[lane];
            float rv;
            if (lane >= 4)      rv = sigmoidf_fast(v);                        // conf/classes
            else if (lane == 0) rv = (sigmoidf_fast(v) + (float)s) * stride;  // cx
            else if (lane == 1) rv = (sigmoidf_fast(v) + fy) * stride;        // cy
            else if (lane == 2) rv = __expf(v) * anchor_w * stride;           // w
            else                rv = __expf(v) * anchor_h * stride;           // h
            __builtin_nontemporal_store(rv, &orow[lane]);
        }
        // Chunk 1: k = 32 + lane (32..63). Pure sigmoid, branch-free.
        {
            const float v = row[32 + lane];
            __builtin_nontemporal_store(sigmoidf_fast(v), &orow[32 + lane]);
        }
        // Chunk 2: k = 64 + lane (64..84). Pure sigmoid, 21 active lanes.
        if (lane < CH - 64) {
            const float v = row[64 + lane];
            __builtin_nontemporal_store(sigmoidf_fast(v), &orow[64 + lane]);
        }
    }
}

extern "C" void kernel_launch(void* const* d_in, const int* in_sizes, int n_in,
                              void* d_out, int out_size, void* d_ws, size_t ws_size,
                              hipStream_t stream) {
    const float* x       = (const float*)d_in[0];
    const int*   img_dim = (const int*)d_in[1];
    float*       out     = (float*)d_out;

    const int B    = in_sizes[0] / (NA * CH * GG);   // 64 for reference shapes
    const int grid = B * NA * G;

    yolo_head_kernel<<<grid, BLOCK, 0, stream>>>(x, img_dim, out);
}